// Router_48515950576546
// MI455X (gfx1250) — compile-verified
//
#include <hip/hip_runtime.h>
#include <cstddef>

typedef __attribute__((ext_vector_type(2))) float v2f;
typedef __attribute__((ext_vector_type(4))) float v4f;
typedef __attribute__((ext_vector_type(8))) float v8f;

// ---------------- problem constants ----------------
#define ROWS      8192          // 4 * 2048
#define DIM       1024
#define H1C       128
#define HS        256
#define POOL      16384
#define TOPK      1024
#define MIN_P     100.0f
#define RANGE_P   924.0f        // MAX_PARAMS - MIN_PARAMS

// output layout (floats) in d_out: indices | weights | mask | complexity
#define OFF_IDX   ((size_t)0)
#define OFF_W     ((size_t)ROWS * TOPK)
#define OFF_MASK  ((size_t)2 * ROWS * TOPK)
#define OFF_COMP  ((size_t)3 * ROWS * TOPK)

// ---------------- shared-mem carving for top-k kernels ----------------
struct TVars { unsigned long long prefix; unsigned selK; unsigned cnt; unsigned pad; };
#define SOFF_CAND  (131072)                 // keys: 16384 * 8
#define SOFF_HIST  (SOFF_CAND + 8192)       // cand: 1024 * 8
#define SOFF_RED   (SOFF_HIST + 1024)       // hist: 256 * 4
#define SOFF_VARS  (SOFF_RED + 1024)        // red : 256 * 4
#define SOFF_XROW  (SOFF_VARS + 16)
#define SOFF_HSROW (SOFF_XROW + 4096)
#define SMEM_TOPK  ((size_t)SOFF_XROW)              // 141328
#define SMEM_FUSED ((size_t)(SOFF_HSROW + 1024))    // 146448

// ---------------- helpers ----------------
__device__ __forceinline__ unsigned mono_from_float(float f) {
    unsigned u = __float_as_uint(f);
    return (u & 0x80000000u) ? ~u : (u | 0x80000000u);
}
__device__ __forceinline__ float float_from_mono(unsigned m) {
    unsigned u = (m & 0x80000000u) ? (m & 0x7FFFFFFFu) : ~m;
    return __uint_as_float(u);
}
__device__ __forceinline__ unsigned long long make_key(float s, unsigned idx) {
    return ((unsigned long long)mono_from_float(s) << 32) | (unsigned)(~idx);
}

// ============================================================
// Kernel 1 (path A): complexity net -> complexity out + budgets in ws
// 128 threads / block, one row per block
// ============================================================
__global__ __launch_bounds__(128) void k_complexity(
    const float* __restrict__ x, const float* __restrict__ w1c,
    const float* __restrict__ b1c, const float* __restrict__ w2c,
    const float* __restrict__ b2c, int* __restrict__ budgets,
    float* __restrict__ out)
{
    __shared__ float xrow[DIM];
    __shared__ float hbuf[H1C];
    const int row = blockIdx.x;
    const int tid = threadIdx.x;
    for (int i = tid; i < DIM; i += 128) xrow[i] = x[(size_t)row * DIM + i];
    __syncthreads();
    float acc = b1c[tid];
    #pragma unroll 8
    for (int d = 0; d < DIM; ++d) acc = fmaf(xrow[d], w1c[(size_t)d * H1C + tid], acc);
    hbuf[tid] = fmaxf(acc, 0.0f) * w2c[tid];
    __syncthreads();
    for (int s = 64; s > 0; s >>= 1) {
        if (tid < s) hbuf[tid] += hbuf[tid + s];
        __syncthreads();
    }
    if (tid == 0) {
        float z = hbuf[0] + b2c[0];
        float c = 1.0f / (1.0f + expf(-z));
        float raw = MIN_P + RANGE_P * c * c;
        raw = fminf(fmaxf(raw, MIN_P), MIN_P + RANGE_P);
        budgets[row] = (int)rintf(raw);
        out[OFF_COMP + row] = c;
    }
}

// ============================================================
// fp32 WMMA GEMM: C = act(A[MxK] @ B[KxN] + bias)
// 128 threads = 4 waves. Tile: 32 rows x 64 cols.
// Each wave: one 16-col strip, two 16x16 accumulators (rows 0-15, 16-31)
// sharing each B fragment (2 WMMAs per B ds_load_b64).
// A staged as [32][68] (row pad -> conflict-free ds_load_2addr_b64).
// B staged k-packed: blds[(k/4)*64 + col][k%4] -> one aligned ds_load_b64
// per lane, all 64 banks hit exactly once per wave (conflict-free).
// Inner loop contains NO global accesses and no 64-bit address math.
// ============================================================
template<bool RELU>
__global__ __launch_bounds__(128) void gemm_wmma_f32(
    const float* __restrict__ A, const float* __restrict__ B,
    const float* __restrict__ bias, float* __restrict__ C,
    int N, int Kdim)
{
    __shared__ float Alds[32][68];
    __shared__ float blds[64 * 64];       // k-chunk 64 x 64 cols, k-packed by 4

    const int row0   = blockIdx.y * 32;
    const int colBlk = blockIdx.x * 64;
    const int tid    = threadIdx.x;
    const int wave   = tid >> 5;
    const int lane   = tid & 31;
    const int n      = lane & 15;         // col within 16-wide strip / row of A tile
    const int half2  = (lane >> 4) * 2;   // lanes 16-31 hold K+2,K+3
    const int col0   = colBlk + wave * 16;
    const int bfrag0 = (wave * 16 + n) * 4 + half2;   // float index into blds

    v8f acc0 = {};
    v8f acc1 = {};
    for (int k0 = 0; k0 < Kdim; k0 += 64) {
        __syncthreads();
        // stage A tile 32 x 64 (coalesced rows)
        #pragma unroll 4
        for (int i = tid; i < 32 * 64; i += 128) {
            int r = i >> 6, c = i & 63;
            Alds[r][c] = A[(size_t)(row0 + r) * Kdim + k0 + c];
        }
        // stage B chunk 64k x 64cols, k-packed (coalesced global rows)
        #pragma unroll 4
        for (int i = tid; i < 64 * 64; i += 128) {
            int k = i >> 6, c = i & 63;
            blds[(((k >> 2) * 64 + c) << 2) + (k & 3)] =
                B[(size_t)(k0 + k) * N + colBlk + c];
        }
        __syncthreads();
        #pragma unroll
        for (int kc = 0; kc < 64; kc += 4) {
            const int kk = kc + half2;
            v2f a0 = *(const v2f*)&Alds[n][kk];
            v2f a1 = *(const v2f*)&Alds[n + 16][kk];
            v2f b  = *(const v2f*)&blds[bfrag0 + (kc << 6)];
            acc0 = __builtin_amdgcn_wmma_f32_16x16x4_f32(
                false, a0, false, b, (short)0, acc0, false, false);
            acc1 = __builtin_amdgcn_wmma_f32_16x16x4_f32(
                false, a1, false, b, (short)0, acc1, false, false);
        }
    }
    const float bv = bias[col0 + n];
    const int   rbase = row0 + (half2 << 2);   // + half*8
    #pragma unroll
    for (int r = 0; r < 8; ++r) {
        float v0 = acc0[r] + bv;
        float v1 = acc1[r] + bv;
        if (RELU) { v0 = fmaxf(v0, 0.0f); v1 = fmaxf(v1, 0.0f); }
        C[(size_t)(rbase + r) * N + col0 + n]      = v0;
        C[(size_t)(rbase + r + 16) * N + col0 + n] = v1;
    }
}

// ============================================================
// Top-k core: keys[16384] (64-bit sortable, tie-exact) -> exact top-1024
// (radix-select threshold, compact, bitonic sort desc), softmax, mask, emit.
// blockDim.x == 256
// ============================================================
__device__ void topk_core(unsigned long long* keys, unsigned long long* cand,
                          unsigned* hist, float* red, TVars* vars,
                          int row, int budget, float* __restrict__ out)
{
    const int tid = threadIdx.x;
    if (tid == 0) { vars->prefix = 0ull; vars->selK = TOPK; vars->cnt = 0u; }
    unsigned long long pmask = 0ull;
    __syncthreads();

    // 8-pass MSB-first radix select on 64-bit keys -> exact 1024th-largest key
    for (int p = 0; p < 8; ++p) {
        const int shift = 56 - 8 * p;
        hist[tid] = 0u;
        __syncthreads();
        const unsigned long long pref = vars->prefix;
        for (int i = tid; i < POOL; i += 256) {
            unsigned long long k = keys[i];
            if ((k & pmask) == pref)
                atomicAdd(&hist[(unsigned)((k >> shift) & 0xFFull)], 1u);
        }
        __syncthreads();
        if (tid == 0) {
            unsigned Kt = vars->selK, cum = 0; int d = 0;
            for (d = 255; d > 0; --d) {
                unsigned h = hist[d];
                if (cum + h >= Kt) break;
                cum += h;
            }
            vars->selK = Kt - cum;
            vars->prefix = pref | ((unsigned long long)(unsigned)d << shift);
        }
        pmask |= (0xFFull << shift);
        __syncthreads();
    }
    const unsigned long long thr = vars->prefix;

    // compact survivors (exactly 1024, keys are unique by construction)
    for (int i = tid; i < POOL; i += 256) {
        unsigned long long k = keys[i];
        if (k >= thr) { unsigned pos = atomicAdd(&vars->cnt, 1u); cand[pos] = k; }
    }
    __syncthreads();

    // bitonic sort 1024 keys, descending
    for (unsigned sz = 2; sz <= TOPK; sz <<= 1) {
        for (unsigned st = sz >> 1; st > 0; st >>= 1) {
            for (unsigned t = tid; t < TOPK; t += 256) {
                unsigned j = t ^ st;
                if (j > t) {
                    unsigned long long a = cand[t], b = cand[j];
                    bool desc = ((t & sz) == 0);
                    if (desc ? (a < b) : (a > b)) { cand[t] = b; cand[j] = a; }
                }
            }
            __syncthreads();
        }
    }

    // softmax over top scores (max = first element), budget mask, emit
    const float maxs = float_from_mono((unsigned)(cand[0] >> 32));
    float part = 0.0f;
    for (int t = tid; t < TOPK; t += 256)
        part += expf(float_from_mono((unsigned)(cand[t] >> 32)) - maxs);
    red[tid] = part;
    __syncthreads();
    for (int s = 128; s > 0; s >>= 1) {
        if (tid < s) red[tid] += red[tid + s];
        __syncthreads();
    }
    const float inv = 1.0f / red[0];
    const size_t base = (size_t)row * TOPK;
    for (int t = tid; t < TOPK; t += 256) {
        unsigned long long k = cand[t];
        unsigned idx = ~((unsigned)k);
        float s = float_from_mono((unsigned)(k >> 32));
        float w = expf(s - maxs) * inv;
        float msk = (t < budget) ? 1.0f : 0.0f;
        out[OFF_IDX  + base + t] = (float)idx;
        out[OFF_W    + base + t] = w * msk;
        out[OFF_MASK + base + t] = msk;
    }
}

// ============================================================
// Kernel 4 (path A): read materialized scores row -> top-k
// ============================================================
__global__ __launch_bounds__(256) void k_topk(
    const float* __restrict__ scores, const int* __restrict__ budgets,
    float* __restrict__ out)
{
    extern __shared__ unsigned char smem[];
    unsigned long long* keys = (unsigned long long*)smem;
    unsigned long long* cand = (unsigned long long*)(smem + SOFF_CAND);
    unsigned* hist = (unsigned*)(smem + SOFF_HIST);
    float*    red  = (float*)(smem + SOFF_RED);
    TVars*    vars = (TVars*)(smem + SOFF_VARS);

    const int row = blockIdx.x;
    const int tid = threadIdx.x;
    const float* srow = scores + (size_t)row * POOL;
    // b128 loads, keys built in LDS
    for (int i = tid; i < POOL / 4; i += 256) {
        v4f v = *(const v4f*)&srow[i * 4];
        keys[i * 4 + 0] = make_key(v.x, (unsigned)(i * 4 + 0));
        keys[i * 4 + 1] = make_key(v.y, (unsigned)(i * 4 + 1));
        keys[i * 4 + 2] = make_key(v.z, (unsigned)(i * 4 + 2));
        keys[i * 4 + 3] = make_key(v.w, (unsigned)(i * 4 + 3));
    }
    const int budget = budgets[row];
    __syncthreads();
    topk_core(keys, cand, hist, red, vars, row, budget, out);
}

// ============================================================
// Fallback (path B): fully fused, zero-workspace, one row per block
// ============================================================
__global__ __launch_bounds__(256) void k_fused_row(
    const float* __restrict__ x,
    const float* __restrict__ w1c, const float* __restrict__ b1c,
    const float* __restrict__ w2c, const float* __restrict__ b2c,
    const float* __restrict__ ws1, const float* __restrict__ bs1,
    const float* __restrict__ ws2, const float* __restrict__ bs2,
    float* __restrict__ out)
{
    extern __shared__ unsigned char smem[];
    unsigned long long* keys = (unsigned long long*)smem;
    unsigned long long* cand = (unsigned long long*)(smem + SOFF_CAND);
    unsigned* hist  = (unsigned*)(smem + SOFF_HIST);
    float*    red   = (float*)(smem + SOFF_RED);
    TVars*    vars  = (TVars*)(smem + SOFF_VARS);
    float*    xrow  = (float*)(smem + SOFF_XROW);
    float*    hsrow = (float*)(smem + SOFF_HSROW);

    const int row = blockIdx.x;
    const int tid = threadIdx.x;
    for (int i = tid; i < DIM; i += 256) xrow[i] = x[(size_t)row * DIM + i];
    __syncthreads();

    // complexity -> budget (all threads compute the same scalar)
    float cpart = 0.0f;
    if (tid < H1C) {
        float a = b1c[tid];
        #pragma unroll 8
        for (int d = 0; d < DIM; ++d) a = fmaf(xrow[d], w1c[(size_t)d * H1C + tid], a);
        cpart = fmaxf(a, 0.0f) * w2c[tid];
    }
    red[tid] = cpart;
    __syncthreads();
    for (int s = 128; s > 0; s >>= 1) {
        if (tid < s) red[tid] += red[tid + s];
        __syncthreads();
    }
    const float z = red[0] + b2c[0];
    const float c = 1.0f / (1.0f + expf(-z));
    float raw = fminf(fmaxf(MIN_P + RANGE_P * c * c, MIN_P), MIN_P + RANGE_P);
    const int budget = (int)rintf(raw);
    if (tid == 0) out[OFF_COMP + row] = c;

    // hs row
    {
        float a = bs1[tid];
        #pragma unroll 8
        for (int d = 0; d < DIM; ++d) a = fmaf(xrow[d], ws1[(size_t)d * HS + tid], a);
        hsrow[tid] = fmaxf(a, 0.0f);
    }
    __syncthreads();

    // scores -> keys
    for (int i = 0; i < POOL / 256; ++i) {
        const int col = tid + (i << 8);
        float a = bs2[col];
        #pragma unroll 8
        for (int h = 0; h < HS; ++h)
            a = fmaf(hsrow[h], ws2[(size_t)h * POOL + col], a);
        keys[col] = make_key(a, (unsigned)col);
    }
    __syncthreads();
    topk_core(keys, cand, hist, red, vars, row, budget, out);
}

// ============================================================
// Launcher
// ============================================================
extern "C" void kernel_launch(void* const* d_in, const int* in_sizes, int n_in,
                              void* d_out, int out_size, void* d_ws, size_t ws_size,
                              hipStream_t stream)
{
    const float* x   = (const float*)d_in[0];
    const float* w1c = (const float*)d_in[1];
    const float* b1c = (const float*)d_in[2];
    const float* w2c = (const float*)d_in[3];
    const float* b2c = (const float*)d_in[4];
    const float* ws1 = (const float*)d_in[5];
    const float* bs1 = (const float*)d_in[6];
    const float* ws2 = (const float*)d_in[7];
    const float* bs2 = (const float*)d_in[8];
    float* out = (float*)d_out;

    // workspace layout (path A): [budgets int32 x8192 | hs f32 8192x256 | scores f32 8192x16384]
    const size_t off_hs     = 65536;
    const size_t off_scores = off_hs + (size_t)ROWS * HS * sizeof(float);
    const size_t need_full  = off_scores + (size_t)ROWS * POOL * sizeof(float);

    (void)hipFuncSetAttribute((const void*)k_topk,
        hipFuncAttributeMaxDynamicSharedMemorySize, (int)SMEM_TOPK);
    (void)hipFuncSetAttribute((const void*)k_fused_row,
        hipFuncAttributeMaxDynamicSharedMemorySize, (int)SMEM_FUSED);

    if (ws_size >= need_full) {
        int*   budgets = (int*)d_ws;
        float* hs      = (float*)((char*)d_ws + off_hs);
        float* scores  = (float*)((char*)d_ws + off_scores);

        k_complexity<<<ROWS, 128, 0, stream>>>(x, w1c, b1c, w2c, b2c, budgets, out);
        // hs = relu(X @ Ws1 + bs1) : M=8192, K=1024, N=256
        gemm_wmma_f32<true><<<dim3(HS / 64, ROWS / 32), 128, 0, stream>>>(
            x, ws1, bs1, hs, HS, DIM);
        // scores = hs @ Ws2 + bs2 : M=8192, K=256, N=16384
        gemm_wmma_f32<false><<<dim3(POOL / 64, ROWS / 32), 128, 0, stream>>>(
            hs, ws2, bs2, scores, POOL, HS);
        k_topk<<<ROWS, 256, SMEM_TOPK, stream>>>(scores, budgets, out);
    } else {
        // zero-workspace fused fallback
        k_fused_row<<<ROWS, 256, SMEM_FUSED, stream>>>(
            x, w1c, b1c, w2c, b2c, ws1, bs1, ws2, bs2, out);
    }
}